// SDFLoss__11905649344615
// MI455X (gfx1250) — compile-verified
//
#include <hip/hip_runtime.h>
#include <math.h>

// ---------------------------------------------------------------------------
// SDF loss for MI455X (gfx1250).  FP32 WMMA 16x16x4 batches the ab.p / ac.p
// dot products for 16-point x 16-triangle tiles; scalar FP32 epilogue does the
// region logic (needs full FP32: thresholds 1e-9 / 1e-12).
// ---------------------------------------------------------------------------

#define GRID 32
#define GM1  31
#define STEP (2.0f / 31.0f)

typedef __attribute__((ext_vector_type(2))) float v2f;
typedef __attribute__((ext_vector_type(8))) float v8f;

__device__ __forceinline__ float sdiv_f(float n, float d) {
  float dd = (fabsf(d) < 1e-12f) ? 1e-12f : d;
  return n / dd;
}

// ---------------- kernel 1: bbox of hull 1 -> center(3), scale(1) ----------
__global__ void bbox_kernel(const float* __restrict__ verts, int V,
                            float* __restrict__ cs) {
  __shared__ float smn[3][256];
  __shared__ float smx[3][256];
  const int tid = threadIdx.x;
  float mn[3] = {3.4e38f, 3.4e38f, 3.4e38f};
  float mx[3] = {-3.4e38f, -3.4e38f, -3.4e38f};
  const float* base = verts + (size_t)V * 3;  // hull index 1
  for (int v = tid; v < V; v += 256) {
#pragma unroll
    for (int k = 0; k < 3; ++k) {
      float x = base[v * 3 + k];
      mn[k] = fminf(mn[k], x);
      mx[k] = fmaxf(mx[k], x);
    }
  }
#pragma unroll
  for (int k = 0; k < 3; ++k) { smn[k][tid] = mn[k]; smx[k][tid] = mx[k]; }
  __syncthreads();
  for (int s = 128; s > 0; s >>= 1) {
    if (tid < s) {
#pragma unroll
      for (int k = 0; k < 3; ++k) {
        smn[k][tid] = fminf(smn[k][tid], smn[k][tid + s]);
        smx[k][tid] = fmaxf(smx[k][tid], smx[k][tid + s]);
      }
    }
    __syncthreads();
  }
  if (tid == 0) {
    float ex = smx[0][0] - smn[0][0];
    float ey = smx[1][0] - smn[1][0];
    float ez = smx[2][0] - smn[2][0];
    float ext = fmaxf(ex, fmaxf(ey, ez));
    cs[0] = 0.5f * (smn[0][0] + smx[0][0]);
    cs[1] = 0.5f * (smn[1][0] + smx[1][0]);
    cs[2] = 0.5f * (smn[2][0] + smx[2][0]);
    cs[3] = 0.6f * ext;  // (1 + 0.2) * 0.5 * maxExtent
  }
}

// ---------------- kernel 2: per-triangle records (hull 1, scaled) ----------
// record (32 floats / 128B):
//  [0..2]=a [3..5]=b [6..8]=c [9..11]=ab [12..14]=ac [15..17]=n(=ab x ac)
//  [18..20]=ab.a ab.b ab.c  [21..23]=ac.a ac.b ac.c  [24]=det [25]=inv
__global__ void tri_kernel(const float* __restrict__ verts,
                           const int* __restrict__ faces, int V, int F,
                           int Fpad, const float* __restrict__ cs,
                           float* __restrict__ tri) {
  int f = blockIdx.x * blockDim.x + threadIdx.x;
  if (f >= Fpad) return;
  float r[32];
#pragma unroll
  for (int i = 0; i < 32; ++i) r[i] = 0.0f;
  if (f < F) {
    const float cx = cs[0], cy = cs[1], cz = cs[2], sc = cs[3];
    const float* base = verts + (size_t)V * 3;  // hull 1
    int i0 = faces[3 * f + 0], i1 = faces[3 * f + 1], i2 = faces[3 * f + 2];
    float ax = (base[3 * i0 + 0] - cx) / sc;
    float ay = (base[3 * i0 + 1] - cy) / sc;
    float az = (base[3 * i0 + 2] - cz) / sc;
    float bx = (base[3 * i1 + 0] - cx) / sc;
    float by = (base[3 * i1 + 1] - cy) / sc;
    float bz = (base[3 * i1 + 2] - cz) / sc;
    float px = (base[3 * i2 + 0] - cx) / sc;
    float py = (base[3 * i2 + 1] - cy) / sc;
    float pz = (base[3 * i2 + 2] - cz) / sc;
    float abx = bx - ax, aby = by - ay, abz = bz - az;
    float acx = px - ax, acy = py - ay, acz = pz - az;
    float nx = aby * acz - abz * acy;
    float ny = abz * acx - abx * acz;
    float nz = abx * acy - aby * acx;
    float det = aby * (-acz) + abz * acy;  // ray dir (1,0,0)
    float inv = sdiv_f(1.0f, det);
    r[0] = ax; r[1] = ay; r[2] = az;
    r[3] = bx; r[4] = by; r[5] = bz;
    r[6] = px; r[7] = py; r[8] = pz;
    r[9] = abx; r[10] = aby; r[11] = abz;
    r[12] = acx; r[13] = acy; r[14] = acz;
    r[15] = nx; r[16] = ny; r[17] = nz;
    r[18] = abx * ax + aby * ay + abz * az;
    r[19] = abx * bx + aby * by + abz * bz;
    r[20] = abx * px + aby * py + abz * pz;
    r[21] = acx * ax + acy * ay + acz * az;
    r[22] = acx * bx + acy * by + acz * bz;
    r[23] = acx * px + acy * py + acz * pz;
    r[24] = det; r[25] = inv;
  }
  float4* out = (float4*)(tri + (size_t)f * 32);
#pragma unroll
  for (int i = 0; i < 8; ++i)
    out[i] = make_float4(r[4 * i], r[4 * i + 1], r[4 * i + 2], r[4 * i + 3]);
}

// ---------------- kernel 3: WMMA SDF grid (hull 1) --------------------------
// One wave = 16 grid points; loop over triangle tiles of 16.
// D = A(16x4 points) x B(4x16 tri-edge) via V_WMMA_F32_16X16X4_F32.
__global__ void __launch_bounds__(128) sdf_grid_kernel(
    const float* __restrict__ tri, int nTriTiles, int F,
    float* __restrict__ phi) {
  const int lane = threadIdx.x & 31;
  const int wave = (blockIdx.x * blockDim.x + threadIdx.x) >> 5;
  const int ptBase = wave << 4;  // 16 points per wave
  const bool hiHalf = lane >= 16;
  const int lo = lane & 15;

  // ---- A matrix (points): lane L holds M=L&15; VGPR0: K0(lo)/K2(hi),
  //      VGPR1: K1(lo)/K3(hi)  ->  (px,py) low half, (pz,0) high half
  {
  }
  const int pA = ptBase + lo;
  const float pxA = -1.0f + (float)(pA & 31) * STEP;
  const float pyA = -1.0f + (float)((pA >> 5) & 31) * STEP;
  const float pzA = -1.0f + (float)(pA >> 10) * STEP;
  v2f Amat;
  Amat[0] = hiHalf ? pzA : pxA;
  Amat[1] = hiHalf ? 0.0f : pyA;

  float minD2[8];
  int hits[8];
#pragma unroll
  for (int r = 0; r < 8; ++r) { minD2[r] = 3.4e38f; hits[r] = 0; }

  const v8f czero = {};

  for (int tt = 0; tt < nTriTiles; ++tt) {
    const int triIdx = (tt << 4) + lo;
    const bool valid = triIdx < F;
    const float4* rp = (const float4*)(tri + (size_t)triIdx * 32);
    const float4 L0 = rp[0], L1 = rp[1], L2 = rp[2], L3 = rp[3];
    const float4 L4 = rp[4], L5 = rp[5], L6 = rp[6];
    const float ax = L0.x, ay = L0.y, az = L0.z, bx = L0.w;
    const float by = L1.x, bz = L1.y, cx = L1.z, cy = L1.w;
    const float cz = L2.x, abx = L2.y, aby = L2.z, abz = L2.w;
    const float acx = L3.x, acy = L3.y, acz = L3.z, nx = L3.w;
    const float ny = L4.x, nz = L4.y, dab_a = L4.z, dab_b = L4.w;
    const float dab_c = L5.x, dac_a = L5.y, dac_b = L5.z, dac_c = L5.w;
    const float det = L6.x, inv = L6.y;

    // ---- B matrices (4x16): column N = lo; same K split as A
    v2f Bab, Bac;
    Bab[0] = hiHalf ? abz : abx;
    Bab[1] = hiHalf ? 0.0f : aby;
    Bac[0] = hiHalf ? acz : acx;
    Bac[1] = hiHalf ? 0.0f : acy;

    // D[M][N] = p_M . ab_N  (and . ac_N)
    v8f Sab = __builtin_amdgcn_wmma_f32_16x16x4_f32(
        false, Amat, false, Bab, (short)0, czero, false, false);
    v8f Sac = __builtin_amdgcn_wmma_f32_16x16x4_f32(
        false, Amat, false, Bac, (short)0, czero, false, false);

#pragma unroll
    for (int r = 0; r < 8; ++r) {
      const int q = ptBase + r + (hiHalf ? 8 : 0);
      const float qx = -1.0f + (float)(q & 31) * STEP;
      const float qy = -1.0f + (float)((q >> 5) & 31) * STEP;
      const float qz = -1.0f + (float)(q >> 10) * STEP;

      const float sab = Sab[r], sac = Sac[r];
      const float d1 = sab - dab_a, d2 = sac - dac_a;
      const float d3 = sab - dab_b, d4 = sac - dac_b;
      const float d5 = sab - dab_c, d6 = sac - dac_c;

      const float vc = d1 * d4 - d3 * d2;
      const float vb = d5 * d2 - d1 * d6;
      const float va = d3 * d6 - d5 * d4;
      const float denom = va + vb + vc;
      const float v_in = sdiv_f(vb, denom);
      const float w_in = sdiv_f(vc, denom);
      float rx = ax + abx * v_in + acx * w_in;
      float ry = ay + aby * v_in + acy * w_in;
      float rz = az + abz * v_in + acz * w_in;
      // edge BC
      {
        const bool c1 = (va <= 0.0f) && (d4 - d3 >= 0.0f) && (d5 - d6 >= 0.0f);
        const float w = sdiv_f(d4 - d3, (d4 - d3) + (d5 - d6));
        rx = c1 ? bx + (cx - bx) * w : rx;
        ry = c1 ? by + (cy - by) * w : ry;
        rz = c1 ? bz + (cz - bz) * w : rz;
      }
      // edge AC
      {
        const bool c2 = (vb <= 0.0f) && (d2 >= 0.0f) && (d6 <= 0.0f);
        const float w = sdiv_f(d2, d2 - d6);
        rx = c2 ? ax + acx * w : rx;
        ry = c2 ? ay + acy * w : ry;
        rz = c2 ? az + acz * w : rz;
      }
      // edge AB
      {
        const bool c3 = (vc <= 0.0f) && (d1 >= 0.0f) && (d3 <= 0.0f);
        const float w = sdiv_f(d1, d1 - d3);
        rx = c3 ? ax + abx * w : rx;
        ry = c3 ? ay + aby * w : ry;
        rz = c3 ? az + abz * w : rz;
      }
      // vertex C / B / A (later overrides win, same order as reference)
      {
        const bool c4 = (d6 >= 0.0f) && (d5 <= d6);
        rx = c4 ? cx : rx; ry = c4 ? cy : ry; rz = c4 ? cz : rz;
        const bool c5 = (d3 >= 0.0f) && (d4 <= d3);
        rx = c5 ? bx : rx; ry = c5 ? by : ry; rz = c5 ? bz : rz;
        const bool c6 = (d1 <= 0.0f) && (d2 <= 0.0f);
        rx = c6 ? ax : rx; ry = c6 ? ay : ry; rz = c6 ? az : rz;
      }
      const float dx = qx - rx, dy = qy - ry, dz = qz - rz;
      const float dd = dx * dx + dy * dy + dz * dz;
      minD2[r] = (valid && dd < minD2[r]) ? dd : minD2[r];

      // inside test: ray (1,0,0) from q
      const float sx = qx - ax, sy = qy - ay, sz = qz - az;
      const float u = (sy * (-acz) + sz * acy) * inv;
      const float vv = (sy * abz - sz * aby) * inv;
      const float t = (nx * sx + ny * sy + nz * sz) * inv;
      const bool hit = (fabsf(det) > 1e-9f) && (u >= 0.0f) && (u <= 1.0f) &&
                       (vv >= 0.0f) && (u + vv <= 1.0f) && (t > 0.0f);
      hits[r] += (hit && valid) ? 1 : 0;
    }
  }

  // reduce over the 16 triangle slots (within each half-wave; matches D layout)
#pragma unroll
  for (int r = 0; r < 8; ++r) {
    float m = minD2[r];
    int h = hits[r];
#pragma unroll
    for (int off = 8; off >= 1; off >>= 1) {
      m = fminf(m, __shfl_xor(m, off, 32));
      h += __shfl_xor(h, off, 32);
    }
    if (lo == 0) {
      const int q = ptBase + r + (hiHalf ? 8 : 0);
      phi[q] = (h & 1) ? sqrtf(fmaxf(m, 0.0f)) : 0.0f;
    }
  }
}

// ---------------- kernel 4: trilinear sample + loss -------------------------
// loss = sum_v trilerp(phi1, (verts[0,v]-c1)/s1) / H^2   (H=2, weights[1]=0)
__global__ void loss_kernel(const float* __restrict__ verts, int V,
                            const float* __restrict__ cs,
                            const float* __restrict__ phi,
                            float* __restrict__ out) {
  __shared__ float red[256];
  const int tid = threadIdx.x;
  const float cx = cs[0], cy = cs[1], cz = cs[2], sc = cs[3];
  float acc = 0.0f;
  for (int v = tid; v < V; v += 256) {
    const float x = (verts[3 * v + 0] - cx) / sc;
    const float y = (verts[3 * v + 1] - cy) / sc;
    const float z = (verts[3 * v + 2] - cz) / sc;
    const float fx = (x + 1.0f) * 0.5f * (float)GM1;
    const float fy = (y + 1.0f) * 0.5f * (float)GM1;
    const float fz = (z + 1.0f) * 0.5f * (float)GM1;
    const float x0f = floorf(fx), y0f = floorf(fy), z0f = floorf(fz);
    const float wx = fx - x0f, wy = fy - y0f, wz = fz - z0f;
    const int x0 = (int)x0f, y0 = (int)y0f, z0 = (int)z0f;
    float s = 0.0f;
#pragma unroll
    for (int k = 0; k < 8; ++k) {
      const int zi = z0 + (k >> 2);
      const int yi = y0 + ((k >> 1) & 1);
      const int xi = x0 + (k & 1);
      const bool ok = (zi >= 0) && (zi <= GM1) && (yi >= 0) && (yi <= GM1) &&
                      (xi >= 0) && (xi <= GM1);
      const int zc = min(max(zi, 0), GM1);
      const int yc = min(max(yi, 0), GM1);
      const int xc = min(max(xi, 0), GM1);
      const float val = phi[(zc << 10) + (yc << 5) + xc];
      const float w = ((k >> 2) ? wz : 1.0f - wz) *
                      (((k >> 1) & 1) ? wy : 1.0f - wy) *
                      ((k & 1) ? wx : 1.0f - wx);
      s += ok ? val * w : 0.0f;
    }
    acc += s;
  }
  red[tid] = acc;
  __syncthreads();
  for (int sft = 128; sft > 0; sft >>= 1) {
    if (tid < sft) red[tid] += red[tid + sft];
    __syncthreads();
  }
  if (tid == 0) out[0] = red[0] * 0.25f;  // / H^2, H = 2
}

// ---------------------------------------------------------------------------
extern "C" void kernel_launch(void* const* d_in, const int* in_sizes, int n_in,
                              void* d_out, int out_size, void* d_ws,
                              size_t ws_size, hipStream_t stream) {
  const float* verts = (const float*)d_in[0];
  const int* faces = (const int*)d_in[1];
  const int V = in_sizes[0] / 6;  // H=2, 3 comps
  const int F = in_sizes[1] / 3;
  const int Fpad = ((F + 15) / 16) * 16;
  const int nTT = Fpad / 16;

  float* w = (float*)d_ws;
  float* cs = w;                    // 4 floats: center.xyz, scale
  float* phi = w + 16;              // 32768 floats (z,y,x)
  float* tri = w + 16 + GRID * GRID * GRID;  // Fpad * 32 floats

  bbox_kernel<<<1, 256, 0, stream>>>(verts, V, cs);
  tri_kernel<<<(Fpad + 255) / 256, 256, 0, stream>>>(verts, faces, V, F, Fpad,
                                                     cs, tri);
  // 32768 points / 16 per wave = 2048 waves; 4 waves (128 thr) per block
  sdf_grid_kernel<<<512, 128, 0, stream>>>(tri, nTT, F, phi);
  loss_kernel<<<1, 256, 0, stream>>>(verts, V, cs, phi, (float*)d_out);
}